// ThermalGNN_22789096472588
// MI455X (gfx1250) — compile-verified
//
#include <hip/hip_runtime.h>

// ThermalGNN (2-layer GraphSAGE + linear risk head) for MI455X / gfx1250.
//
// Strategy (memory-bound workload; ~4 GFLOP of GEMM is trivial):
//  * transform-before-aggregate: segment_mean(x[src]) @ W == segment_mean((x@W)[src]),
//    halving the per-edge gather/scatter width (128->64, 64->32).
//  * node tables (<=25.6 MB each) are L2-resident (192 MB L2), so edge gathers and
//    float atomics resolve in L2, not HBM (23.3 TB/s is nowhere near the limiter).
//  * GEMMs use V_WMMA_F32_16X16X4_F32 (fp32-exact, matches fp32 reference).
//    Each wave32 computes a 16 x (16*NT) output band: A loaded once per k-step,
//    NT independent WMMA accumulators per step (no D->A/B hazards).
//    B is staged TRANSPOSED in LDS with a +2-dword padded row stride, so each
//    lane's B operand is one conflict-free ds_load_b64 into an aligned VGPR pair
//    (no repacking movs), and the vector-memory path stays free for streaming A.
//  * risk head: one wave per node, lane==feature, __shfl_xor wave32 reduction.

typedef float v2f __attribute__((ext_vector_type(2)));
typedef float v8f __attribute__((ext_vector_type(8)));

#define GNN_N_NODES 100000
#define GNN_N_EDGES 1600000
#define GNN_IN_F 128
#define GNN_HID 64
#define GNN_EMB 32

// ---------------------------------------------------------------------------
// C[M,N] = A[M,K] @ B[K,N], fp32, via V_WMMA_F32_16X16X4_F32.
// One 16 x (16*NT) output band per wave32 (NT*16 == N). Requires M % 16 == 0,
// K % 4 == 0 (true here: M=100000=16*6250, N in {64,32}, K in {128,64}).
// Dynamic LDS: N*(K+2) floats holding B transposed, padded stride K+2
// (K % 64 == 0 here, so stride ≡ 2 mod 64 banks -> conflict-free b64 reads).
//
// VGPR layouts (CDNA5 ISA 7.12.2):
//  A 16x4:  lanes 0-15 hold M=lane, {K, K+1}; lanes 16-31 hold M=lane-16, {K+2, K+3}
//           -> contiguous 2-float (b64) load per lane per step.
//  B 4x16:  N = lane&15 across lanes, K-halves mirrored like A
//           -> with B^T in LDS this is one contiguous ds_load_b64 per lane.
//  C/D:     8 VGPRs; VGPR r holds M = r + 8*(lane>=16), N = lane&15.
// ---------------------------------------------------------------------------
template <int NT>
__global__ void gnn_gemm_f32_wmma(const float* __restrict__ A,
                                  const float* __restrict__ B,
                                  float* __restrict__ C,
                                  int M, int K, int N) {
  extern __shared__ float ldsBT[];  // [N][K+2]: element (n,k) at n*(K+2)+k
  const int ldk = K + 2;

  // Cooperative transpose-fill: consecutive threads read consecutive n
  // (coalesced global b32), scatter into padded LDS rows.
  for (int i = threadIdx.x; i < K * N; i += blockDim.x) {
    int k = i / N;
    int n = i - k * N;
    ldsBT[n * ldk + k] = B[i];  // B[k*N + n]
  }
  __syncthreads();

  const int lane = threadIdx.x & 31;
  const int wave = threadIdx.x >> 5;
  const int mt = blockIdx.x * (blockDim.x >> 5) + wave;
  if (mt >= (M >> 4)) return;

  const int tm = mt << 4;
  const int r16 = lane & 15;
  const int half = lane >> 4;

  const float* arow = A + (size_t)(tm + r16) * K + 2 * half;  // A[m][2*half + k]

  // Per-tile B^T row pointer: lane reads {B[k+2h][n], B[k+2h+1][n]} as one b64.
  const float* btrow[NT];
#pragma unroll
  for (int t = 0; t < NT; ++t) btrow[t] = ldsBT + (t * 16 + r16) * ldk + 2 * half;

  v8f acc[NT];
#pragma unroll
  for (int t = 0; t < NT; ++t) acc[t] = (v8f){};

  for (int k = 0; k < K; k += 4) {
    v2f a = *(const v2f*)(arow + k);  // global b64: A[m][k+2h], A[m][k+2h+1]
#pragma unroll
    for (int t = 0; t < NT; ++t) {
      v2f b = *(const v2f*)(btrow[t] + k);  // ds_load_b64, conflict-free
      acc[t] = __builtin_amdgcn_wmma_f32_16x16x4_f32(false, a, false, b,
                                                     (short)0, acc[t],
                                                     false, false);
    }
  }

  float* crow = C + (size_t)(tm + 8 * half) * N + r16;
#pragma unroll
  for (int t = 0; t < NT; ++t)
#pragma unroll
    for (int r = 0; r < 8; ++r)
      crow[(size_t)r * N + t * 16] = acc[t][r];
}

// ---------------------------------------------------------------------------
__global__ void gnn_zero_f32(float* __restrict__ p, long long n) {
  long long t = (long long)blockIdx.x * blockDim.x + threadIdx.x;
  if (t < n) p[t] = 0.0f;
}

__global__ void gnn_degree_count(const int* __restrict__ dst,
                                 float* __restrict__ cnt, int nEdges) {
  int t = blockIdx.x * blockDim.x + threadIdx.x;
  if (t < nEdges) atomicAdd(&cnt[dst[t]], 1.0f);
}

// agg[dst[e]*F + 4j .. +3] += feat[src[e]*F + 4j .. +3]; one float4 per thread.
// logFv = log2(F/4). Atomics land in L2 (agg table is L2-resident).
__global__ void gnn_scatter_add4(const float* __restrict__ feat,
                                 const int* __restrict__ src,
                                 const int* __restrict__ dst,
                                 float* __restrict__ agg,
                                 int nEdges, int logFv, int F) {
  long long t = (long long)blockIdx.x * blockDim.x + threadIdx.x;
  long long total = (long long)nEdges << logFv;
  if (t >= total) return;
  int e = (int)(t >> logFv);
  int j = (int)(t & ((1 << logFv) - 1));
  int s = src[e];
  int d = dst[e];
  float4 v = *(const float4*)(feat + (size_t)s * F + 4 * j);
  float* o = agg + (size_t)d * F + 4 * j;
  atomicAdd(o + 0, v.x);
  atomicAdd(o + 1, v.y);
  atomicAdd(o + 2, v.z);
  atomicAdd(o + 3, v.w);
}

// out = relu(agg / max(cnt,1) + bias + root); elementwise, safe in-place (out==root).
__global__ void gnn_mean_bias_relu(const float* __restrict__ agg,
                                   const float* __restrict__ cnt,
                                   const float* __restrict__ bias,
                                   const float* __restrict__ root,
                                   float* __restrict__ out,
                                   int F, int logF, long long total) {
  long long t = (long long)blockIdx.x * blockDim.x + threadIdx.x;
  if (t >= total) return;
  int n = (int)(t >> logF);
  int f = (int)(t & (F - 1));
  float c = cnt[n];
  c = c > 1.0f ? c : 1.0f;
  float v = agg[t] / c + bias[f] + root[t];
  out[t] = v > 0.0f ? v : 0.0f;
}

// One wave32 per node: lane = embedding feature (EMB==32).
// emb = relu(agg/max(cnt,1) + b2 + root); risk = emb . Wh + bh via shfl reduction.
__global__ void gnn_emb_risk(const float* __restrict__ agg,
                             const float* __restrict__ cnt,
                             const float* __restrict__ b2,
                             const float* __restrict__ root,
                             const float* __restrict__ Wh,
                             const float* __restrict__ bh,
                             float* __restrict__ emb,
                             float* __restrict__ risk) {
  int lane = threadIdx.x & 31;
  int node = blockIdx.x * (blockDim.x >> 5) + (threadIdx.x >> 5);
  if (node >= GNN_N_NODES) return;
  size_t idx = (size_t)node * GNN_EMB + lane;
  float c = cnt[node];
  c = c > 1.0f ? c : 1.0f;
  float v = agg[idx] / c + b2[lane] + root[idx];
  v = v > 0.0f ? v : 0.0f;
  emb[idx] = v;
  float p = v * Wh[lane];
#pragma unroll
  for (int off = 16; off > 0; off >>= 1) p += __shfl_xor(p, off, 32);
  if (lane == 0) risk[node] = p + bh[0];
}

// ---------------------------------------------------------------------------
extern "C" void kernel_launch(void* const* d_in, const int* in_sizes, int n_in,
                              void* d_out, int out_size, void* d_ws, size_t ws_size,
                              hipStream_t stream) {
  (void)in_sizes; (void)n_in; (void)out_size; (void)ws_size;

  const float* x   = (const float*)d_in[0];
  const int*   ei  = (const int*)d_in[1];    // [2, N_EDGES]: row0=src, row1=dst
  const float* W1l = (const float*)d_in[2];
  const float* b1  = (const float*)d_in[3];
  const float* W1r = (const float*)d_in[4];
  const float* W2l = (const float*)d_in[5];
  const float* b2  = (const float*)d_in[6];
  const float* W2r = (const float*)d_in[7];
  const float* Wh  = (const float*)d_in[8];
  const float* bh  = (const float*)d_in[9];
  const int* src = ei;
  const int* dst = ei + GNN_N_EDGES;

  // Workspace layout (floats):
  //  cnt  : N_NODES (padded to 256)
  //  bufA : N_NODES*HID   -- xl = x@W1_l; later reused as hl | hr (2 * N*EMB = N*HID)
  //  bufB : N_NODES*HID   -- xr = x@W1_r; overwritten in place by h
  //  bufC : N_NODES*HID   -- agg1; first N*EMB reused as agg2
  float* cnt  = (float*)d_ws;
  float* bufA = cnt + ((GNN_N_NODES + 255) & ~255);
  float* bufB = bufA + (size_t)GNN_N_NODES * GNN_HID;
  float* bufC = bufB + (size_t)GNN_N_NODES * GNN_HID;
  float* hl   = bufA;
  float* hr   = bufA + (size_t)GNN_N_NODES * GNN_EMB;
  float* emb  = (float*)d_out;
  float* risk = emb + (size_t)GNN_N_NODES * GNN_EMB;

  const int TPB = 256;  // 8 wave32 per block

  // ---- degree counts (shared by both layers) + zero agg1 ----
  {
    long long n = (long long)GNN_N_NODES * GNN_HID;
    gnn_zero_f32<<<(unsigned)((n + TPB - 1) / TPB), TPB, 0, stream>>>(bufC, n);
    gnn_zero_f32<<<(GNN_N_NODES + TPB - 1) / TPB, TPB, 0, stream>>>(cnt, GNN_N_NODES);
    gnn_degree_count<<<(GNN_N_EDGES + TPB - 1) / TPB, TPB, 0, stream>>>(dst, cnt, GNN_N_EDGES);
  }

  // ---- layer 1: transform (WMMA, 16x64 band/wave), then aggregate 64-wide ----
  {
    int mtiles = GNN_N_NODES / 16;            // 6250
    int blocks = (mtiles + 7) / 8;
    size_t ldsBytes = (size_t)GNN_HID * (GNN_IN_F + 2) * sizeof(float);  // 32.5 KB
    gnn_gemm_f32_wmma<4><<<blocks, TPB, ldsBytes, stream>>>(
        x, W1l, bufA, GNN_N_NODES, GNN_IN_F, GNN_HID);
    gnn_gemm_f32_wmma<4><<<blocks, TPB, ldsBytes, stream>>>(
        x, W1r, bufB, GNN_N_NODES, GNN_IN_F, GNN_HID);

    long long total = (long long)GNN_N_EDGES * (GNN_HID / 4);
    gnn_scatter_add4<<<(unsigned)((total + TPB - 1) / TPB), TPB, 0, stream>>>(
        bufA, src, dst, bufC, GNN_N_EDGES, /*logFv=*/4, GNN_HID);

    long long tot = (long long)GNN_N_NODES * GNN_HID;
    gnn_mean_bias_relu<<<(unsigned)((tot + TPB - 1) / TPB), TPB, 0, stream>>>(
        bufC, cnt, b1, bufB, bufB, GNN_HID, /*logF=*/6, tot);  // h in place over xr
  }

  // ---- layer 2: transform (WMMA, 16x32 band/wave), then aggregate 32-wide ----
  {
    int mtiles = GNN_N_NODES / 16;
    int blocks = (mtiles + 7) / 8;
    size_t ldsBytes = (size_t)GNN_EMB * (GNN_HID + 2) * sizeof(float);  // 8.25 KB
    gnn_gemm_f32_wmma<2><<<blocks, TPB, ldsBytes, stream>>>(
        bufB, W2l, hl, GNN_N_NODES, GNN_HID, GNN_EMB);
    gnn_gemm_f32_wmma<2><<<blocks, TPB, ldsBytes, stream>>>(
        bufB, W2r, hr, GNN_N_NODES, GNN_HID, GNN_EMB);

    long long n = (long long)GNN_N_NODES * GNN_EMB;
    gnn_zero_f32<<<(unsigned)((n + TPB - 1) / TPB), TPB, 0, stream>>>(bufC, n);

    long long total = (long long)GNN_N_EDGES * (GNN_EMB / 4);
    gnn_scatter_add4<<<(unsigned)((total + TPB - 1) / TPB), TPB, 0, stream>>>(
        hl, src, dst, bufC, GNN_N_EDGES, /*logFv=*/3, GNN_EMB);
  }

  // ---- fused embedding + risk head: one wave32 per node ----
  gnn_emb_risk<<<(GNN_N_NODES + 3) / 4, 128, 0, stream>>>(bufC, cnt, b2, hr, Wh, bh, emb, risk);
}